// TemporalRoIHeads_7361573945360
// MI455X (gfx1250) — compile-verified
//
// CDNA5 / gfx1250 — RoI-heads postprocess (softmax + box decode + filter +
// deterministic top-K-by-histogram + greedy batched NMS).
// Memory-bound (~60MB in -> ~2.6us floor @ 23.3 TB/s). No matmul exists in this
// workload, so no WMMA; CDNA5-specific paths used instead:
//   * global_load_async_to_lds_* staging of the NMS working set into LDS
//     (ASYNCcnt + s_wait_asynccnt)
//   * global_prefetch_b8 on the streaming logits reads
//   * wave32-width shuffle reductions (gfx1250 is wave32-only)

#include <hip/hip_runtime.h>
#include <math.h>
#include <stdint.h>

#define NB      8        // batch
#define NP      4096     // proposals per image
#define NC      91       // classes (incl. background)
#define NBINS   4096     // score histogram bins
#define CAP     2048     // candidate budget (matches reference TOPK)
#define DETS    100      // detections per image
#define SCORE_TH 0.05f
#define NMS_TH   0.5f
#define MIN_SZ   0.01f
#define XCLIP    4.135166556742356f   // log(1000/16)

#define NMS_THREADS 1024

#if defined(__has_builtin)
#if __has_builtin(__builtin_amdgcn_global_load_async_to_lds_b32) && \
    __has_builtin(__builtin_amdgcn_global_load_async_to_lds_b128)
#define USE_ASYNC_LDS 1
#endif
#endif
#ifndef USE_ASYNC_LDS
#define USE_ASYNC_LDS 0
#endif

#define AS1 __attribute__((address_space(1)))
#define AS3 __attribute__((address_space(3)))
typedef int v4i_ __attribute__((vector_size(16)));   // matches builtin's pointee

// ---------------- workspace layout (bytes) ----------------
#define WS_HIST    0u                                   // int  [NB][NBINS]
#define WS_TBIN    (WS_HIST + NB*NBINS*4u)              // int  [NB]
#define WS_ROWSTAT (WS_TBIN + 32u)                      // float2 [NB*NP]
#define WS_CCOUNT  (WS_ROWSTAT + NB*NP*8u)              // int  [NB]
#define WS_CSCORE  (WS_CCOUNT + 32u)                    // float[NB][CAP]
#define WS_CBOX    (WS_CSCORE + NB*CAP*4u)              // float4[NB][CAP]
#define WS_CLABEL  (WS_CBOX + NB*CAP*16u)               // int  [NB][CAP]

// ---------------- shared device helpers ----------------
__device__ __forceinline__ int score_bin(float sc) {
  int b = (int)((sc - SCORE_TH) * ((float)NBINS / (1.0f - SCORE_TH)));
  b = b < 0 ? 0 : b;
  return b > (NBINS - 1) ? (NBINS - 1) : b;
}

__device__ __forceinline__ bool decode_clip(float4 r, float pw, float ph,
                                            float cx, float cy,
                                            float Wf, float Hf, float4* out) {
  float dx = r.x * 0.1f, dy = r.y * 0.1f;                 // /wx, /wy (w=10)
  float dw = fminf(r.z * 0.2f, XCLIP);                    // /ww (w=5), clipped
  float dh = fminf(r.w * 0.2f, XCLIP);
  float px = dx * pw + cx, py = dy * ph + cy;
  float qw = __expf(dw) * pw, qh = __expf(dh) * ph;
  float x1 = px - 0.5f * qw, y1 = py - 0.5f * qh;
  float x2 = px + 0.5f * qw, y2 = py + 0.5f * qh;
  x1 = fminf(fmaxf(x1, 0.0f), Wf);  x2 = fminf(fmaxf(x2, 0.0f), Wf);
  y1 = fminf(fmaxf(y1, 0.0f), Hf);  y2 = fminf(fmaxf(y2, 0.0f), Hf);
  *out = make_float4(x1, y1, x2, y2);
  return (x2 - x1 >= MIN_SZ) && (y2 - y1 >= MIN_SZ);
}

// ---------------- kernel 0: zero histogram/counters ----------------
__global__ void k_zero(int* hist, int* ccount, int* tbin) {
  int i = blockIdx.x * blockDim.x + threadIdx.x;
  if (i < NB * NBINS) hist[i] = 0;
  if (i < NB) { ccount[i] = 0; tbin[i] = 0; }
}

// ---------------- kernel 1: softmax stats + validity histogram ----------------
// one wave32 per proposal; lane covers classes {lane, lane+32, lane+64}
__global__ __launch_bounds__(256)
void k_pass1(const float* __restrict__ logits, const float* __restrict__ reg,
             const float* __restrict__ props,  const int* __restrict__ shapes,
             float2* __restrict__ rowstat, int* __restrict__ hist) {
  int wave = threadIdx.x >> 5, lane = threadIdx.x & 31;
  int p = blockIdx.x * 8 + wave;                 // proposal index in [0, NB*NP)
  int img = p >> 12;
  const float* lrow = logits + (size_t)p * NC;
  __builtin_prefetch(lrow + 8 * NC, 0, 1);       // gfx1250: global_prefetch_b8
  float l0 = (lane      < NC) ? lrow[lane]      : -3.0e38f;
  float l1 = (lane + 32 < NC) ? lrow[lane + 32] : -3.0e38f;
  float l2 = (lane + 64 < NC) ? lrow[lane + 64] : -3.0e38f;
  float m = fmaxf(l0, fmaxf(l1, l2));
  for (int o = 16; o; o >>= 1) m = fmaxf(m, __shfl_xor(m, o, 32));
  float e0 = __expf(l0 - m), e1 = __expf(l1 - m), e2 = __expf(l2 - m);
  float s = e0 + e1 + e2;
  for (int o = 16; o; o >>= 1) s += __shfl_xor(s, o, 32);
  if (lane == 0) rowstat[p] = make_float2(m, s);
  float inv = 1.0f / s;

  float4 pr = reinterpret_cast<const float4*>(props)[p];
  float pw = pr.z - pr.x, ph = pr.w - pr.y;
  float cx = pr.x + 0.5f * pw, cy = pr.y + 0.5f * ph;
  float Hf = (float)shapes[img * 2 + 0], Wf = (float)shapes[img * 2 + 1];
  const float* rrow = reg + (size_t)p * (4 * NC);

  float e[3] = {e0, e1, e2};
  for (int k = 0; k < 3; ++k) {
    int c = lane + 32 * k;
    if (c < 1 || c >= NC) continue;              // skip background + OOB
    float sc = e[k] * inv;
    if (sc <= SCORE_TH) continue;
    float4 r = *reinterpret_cast<const float4*>(rrow + 4 * c);
    float4 box;
    if (!decode_clip(r, pw, ph, cx, cy, Wf, Hf, &box)) continue;
    atomicAdd(&hist[img * NBINS + score_bin(sc)], 1);
  }
}

// ---------------- kernel 2: deterministic top-CAP threshold bin ----------------
__global__ void k_thresh(const int* __restrict__ hist, int* __restrict__ tbin) {
  int img = threadIdx.x;
  if (img >= NB) return;
  const int* h = hist + img * NBINS;
  int acc = 0, t = NBINS;
  for (int b = NBINS - 1; b >= 0; --b) {
    if (acc + h[b] > CAP) break;
    acc += h[b]; t = b;
  }
  tbin[img] = t;
}

// ---------------- kernel 3: compact candidates above threshold ----------------
__global__ __launch_bounds__(256)
void k_pass2(const float* __restrict__ logits, const float* __restrict__ reg,
             const float* __restrict__ props,  const int* __restrict__ shapes,
             const float2* __restrict__ rowstat, const int* __restrict__ tbin,
             int* __restrict__ ccount, float* __restrict__ cscore,
             float4* __restrict__ cbox, int* __restrict__ clabel) {
  int wave = threadIdx.x >> 5, lane = threadIdx.x & 31;
  int p = blockIdx.x * 8 + wave;
  int img = p >> 12;
  const float* lrow = logits + (size_t)p * NC;
  __builtin_prefetch(lrow + 8 * NC, 0, 1);
  float2 rs = rowstat[p];
  float m = rs.x, inv = 1.0f / rs.y;
  int t = tbin[img];

  float4 pr = reinterpret_cast<const float4*>(props)[p];
  float pw = pr.z - pr.x, ph = pr.w - pr.y;
  float cx = pr.x + 0.5f * pw, cy = pr.y + 0.5f * ph;
  float Hf = (float)shapes[img * 2 + 0], Wf = (float)shapes[img * 2 + 1];
  const float* rrow = reg + (size_t)p * (4 * NC);

  for (int k = 0; k < 3; ++k) {
    int c = lane + 32 * k;
    if (c < 1 || c >= NC) continue;
    float l = lrow[c];
    float sc = __expf(l - m) * inv;               // bit-identical to pass1 math
    if (sc <= SCORE_TH) continue;
    float4 r = *reinterpret_cast<const float4*>(rrow + 4 * c);
    float4 box;
    if (!decode_clip(r, pw, ph, cx, cy, Wf, Hf, &box)) continue;
    if (score_bin(sc) < t) continue;              // deterministic membership
    int idx = atomicAdd(&ccount[img], 1);
    if (idx < CAP) {
      cscore[img * CAP + idx] = sc;
      cbox[(size_t)img * CAP + idx] = box;
      clabel[img * CAP + idx] = c;
    }
  }
}

// ---------------- kernel 4: greedy NMS fully in LDS ----------------
__global__ __launch_bounds__(NMS_THREADS)
void k_nms(const float* __restrict__ cscore, const float4* __restrict__ cbox,
           const int* __restrict__ clabel, const int* __restrict__ ccount,
           float* __restrict__ out) {
  __shared__ float  s_score[CAP];       // 8 KB
  __shared__ float4 s_box[CAP];         // 32 KB
  __shared__ int    s_label[CAP];       // 8 KB
  __shared__ float  s_rb[NMS_THREADS / 32];
  __shared__ int    s_ri[NMS_THREADS / 32];
  __shared__ int    s_keepi[DETS];
  __shared__ float  s_keeps[DETS];
  __shared__ float  s_bx[4];
  __shared__ float  s_barea;
  __shared__ int    s_bl;

  int img = blockIdx.x, tid = threadIdx.x;
  int cnt = ccount[img];
  cnt = cnt > CAP ? CAP : cnt;
  const float*  gs = cscore + img * CAP;
  const float4* gb = cbox + (size_t)img * CAP;
  const int*    gl = clabel + img * CAP;

  // Stage the whole NMS working set into LDS.
#if USE_ASYNC_LDS
  for (int i = tid; i < CAP; i += NMS_THREADS) {
    __builtin_amdgcn_global_load_async_to_lds_b32(
        (AS1 int*)(uintptr_t)(gs + i),
        (AS3 int*)(AS3 void*)&s_score[i], 0, 0);
    __builtin_amdgcn_global_load_async_to_lds_b128(
        (AS1 v4i_*)(uintptr_t)(gb + i),
        (AS3 v4i_*)(AS3 void*)&s_box[i], 0, 0);
    __builtin_amdgcn_global_load_async_to_lds_b32(
        (AS1 int*)(uintptr_t)(gl + i),
        (AS3 int*)(AS3 void*)&s_label[i], 0, 0);
  }
  asm volatile("s_wait_asynccnt 0" ::: "memory");
#else
  for (int i = tid; i < CAP; i += NMS_THREADS) {
    s_score[i] = gs[i];
    s_box[i]   = gb[i];
    s_label[i] = gl[i];
  }
#endif
  __syncthreads();
  for (int i = tid; i < CAP; i += NMS_THREADS)
    if (i >= cnt) s_score[i] = -1.0f;             // mask padding (reference -1)
  __syncthreads();

  for (int it = 0; it < DETS; ++it) {
    // two-level argmax, lowest-index tie-break (matches jnp.argmax)
    float bs = -2.0f; int bi = CAP;
    for (int j = tid; j < CAP; j += NMS_THREADS) {
      float v = s_score[j];
      if (v > bs || (v == bs && j < bi)) { bs = v; bi = j; }
    }
    for (int o = 16; o; o >>= 1) {                 // wave32 shuffle reduce
      float ov = __shfl_xor(bs, o, 32);
      int   oi = __shfl_xor(bi, o, 32);
      if (ov > bs || (ov == bs && oi < bi)) { bs = ov; bi = oi; }
    }
    if ((tid & 31) == 0) { s_rb[tid >> 5] = bs; s_ri[tid >> 5] = bi; }
    __syncthreads();
    if (tid == 0) {
      float fb = s_rb[0]; int fi = s_ri[0];
      for (int k = 1; k < NMS_THREADS / 32; ++k)
        if (s_rb[k] > fb || (s_rb[k] == fb && s_ri[k] < fi)) { fb = s_rb[k]; fi = s_ri[k]; }
      if (fi >= CAP) fi = 0;
      s_keepi[it] = fi; s_keeps[it] = fb;
      float4 b = s_box[fi];
      s_bx[0] = b.x; s_bx[1] = b.y; s_bx[2] = b.z; s_bx[3] = b.w;
      s_barea = (b.z - b.x) * (b.w - b.y);
      s_bl = s_label[fi];
      s_score[fi] = -1.0f;
    }
    __syncthreads();
    if (s_keeps[it] > 0.0f) {
      float bx1 = s_bx[0], by1 = s_bx[1], bx2 = s_bx[2], by2 = s_bx[3];
      float ba = s_barea; int bl = s_bl;
      // same-label IoU suppression == reference's +10000*label offset trick
      for (int j = tid; j < CAP; j += NMS_THREADS) {
        if (s_score[j] < 0.0f || s_label[j] != bl) continue;
        float4 c = s_box[j];
        float ix1 = fmaxf(bx1, c.x), iy1 = fmaxf(by1, c.y);
        float ix2 = fminf(bx2, c.z), iy2 = fminf(by2, c.w);
        float iw = fmaxf(ix2 - ix1, 0.0f), ih = fmaxf(iy2 - iy1, 0.0f);
        float inter = iw * ih;
        float ca = (c.z - c.x) * (c.w - c.y);
        float iou = inter / (ba + ca - inter);
        if (iou > NMS_TH) s_score[j] = -1.0f;
      }
    }
    __syncthreads();
  }

  // outputs: boxes [8][100][4] | scores [8][100] | labels [8][100] (flat)
  if (tid < DETS) {
    int i = s_keepi[tid]; float s = s_keeps[tid];
    bool ok = s > 0.0f;
    float4 b = ok ? s_box[i] : make_float4(0.f, 0.f, 0.f, 0.f);
    float lab = ok ? (float)s_label[i] : 0.0f;
    float* ob = out + (size_t)img * DETS * 4;
    ob[tid * 4 + 0] = b.x; ob[tid * 4 + 1] = b.y;
    ob[tid * 4 + 2] = b.z; ob[tid * 4 + 3] = b.w;
    out[NB * DETS * 4 + img * DETS + tid] = ok ? s : 0.0f;
    out[NB * DETS * 4 + NB * DETS + img * DETS + tid] = lab;
  }
}

// ---------------- host entry ----------------
extern "C" void kernel_launch(void* const* d_in, const int* in_sizes, int n_in,
                              void* d_out, int out_size, void* d_ws, size_t ws_size,
                              hipStream_t stream) {
  (void)in_sizes; (void)n_in; (void)out_size; (void)ws_size;
  const float* logits = (const float*)d_in[0];   // [8,4096,91]
  const float* reg    = (const float*)d_in[1];   // [8,4096,364]
  const float* props  = (const float*)d_in[2];   // [8,4096,4]
  const int*   shapes = (const int*)d_in[3];     // [8,2] (H,W)

  char* ws = (char*)d_ws;
  int*    hist    = (int*)(ws + WS_HIST);
  int*    tbin    = (int*)(ws + WS_TBIN);
  float2* rowstat = (float2*)(ws + WS_ROWSTAT);
  int*    ccount  = (int*)(ws + WS_CCOUNT);
  float*  cscore  = (float*)(ws + WS_CSCORE);
  float4* cbox    = (float4*)(ws + WS_CBOX);
  int*    clabel  = (int*)(ws + WS_CLABEL);

  k_zero<<<(NB * NBINS + 255) / 256, 256, 0, stream>>>(hist, ccount, tbin);
  k_pass1<<<NB * NP / 8, 256, 0, stream>>>(logits, reg, props, shapes, rowstat, hist);
  k_thresh<<<1, 32, 0, stream>>>(hist, tbin);
  k_pass2<<<NB * NP / 8, 256, 0, stream>>>(logits, reg, props, shapes, rowstat, tbin,
                                           ccount, cscore, cbox, clabel);
  k_nms<<<NB, NMS_THREADS, 0, stream>>>(cscore, cbox, clabel, ccount, (float*)d_out);
}